// GAT_37108517437515
// MI455X (gfx1250) — compile-verified
//
#include <hip/hip_runtime.h>
#include <cstddef>

typedef float v2f __attribute__((ext_vector_type(2)));
typedef float v8f __attribute__((ext_vector_type(8)));

#define HC 256      // H*C
#define CC 64       // C
#define NH 4        // heads
#define TILE_K 64
#define NEG_SLOPE 0.2f

// ---------------- utility ----------------
__global__ void gat_zero_f(float* __restrict__ p, size_t n) {
    size_t i = (size_t)blockIdx.x * blockDim.x + threadIdx.x;
    if (i < n) p[i] = 0.0f;
}

__device__ __forceinline__ unsigned enc_f(float f) {
    unsigned u = __float_as_uint(f);
    return (u & 0x80000000u) ? ~u : (u | 0x80000000u);
}
__device__ __forceinline__ float dec_f(unsigned e) {
    return (e & 0x80000000u) ? __uint_as_float(e ^ 0x80000000u)
                             : __uint_as_float(~e);
}

// ---------------- fp32 WMMA GEMM: Hout[N,OUT] = X[N,K] @ W[K,OUT] ----------------
// block = 256 threads = 8 waves; each wave owns one 16x16 output tile.
// blockIdx.x -> 16-node row tile, blockIdx.y -> 128-column group (wave picks 16 cols).
__global__ __launch_bounds__(256) void gat_gemm_wmma(
        const float* __restrict__ X, const float* __restrict__ W,
        float* __restrict__ Hout, int K, int OUT, int N) {
    __shared__ float As[16][TILE_K + 1];
    const int wave = threadIdx.x >> 5;
    const int lane = threadIdx.x & 31;
    const int grp  = lane >> 4;     // 0/1 -> K-pair select (A/B), row-half (C/D)
    const int ln   = lane & 15;     // m for A, n for B/C/D
    const int row0 = blockIdx.x * 16;
    const int col0 = blockIdx.y * 128 + wave * 16;

    v8f acc = {};
    for (int kb = 0; kb < K; kb += TILE_K) {
        // cooperative stage of 16 x 64 A-tile (1024 floats / 256 threads, float4 each)
        {
            int r  = threadIdx.x >> 4;       // 0..15
            int c4 = (threadIdx.x & 15) * 4; // 0..60
            int rr = row0 + r; if (rr >= N) rr = N - 1;
            const float4 xv = *(const float4*)(X + (size_t)rr * K + kb + c4);
            As[r][c4 + 0] = xv.x; As[r][c4 + 1] = xv.y;
            As[r][c4 + 2] = xv.z; As[r][c4 + 3] = xv.w;
        }
        __syncthreads();
        #pragma unroll
        for (int k = 0; k < TILE_K; k += 4) {
            v2f a, b;
            a.x = As[ln][k + grp * 2 + 0];
            a.y = As[ln][k + grp * 2 + 1];
            const float* Wp = W + (size_t)(kb + k + grp * 2) * OUT + col0 + ln;
            b.x = Wp[0];
            b.y = Wp[OUT];
            acc = __builtin_amdgcn_wmma_f32_16x16x4_f32(
                false, a, false, b, (short)0, acc, false, false);
        }
        __syncthreads();
    }
    #pragma unroll
    for (int j = 0; j < 8; ++j) {
        int r = row0 + j + grp * 8;
        if (r < N) Hout[(size_t)r * OUT + col0 + ln] = acc[j];
    }
}

// ---------------- per-node attention dots: al = sum_c h[n,h,c]*a[h,c] ----------------
__global__ __launch_bounds__(256) void gat_attn_dots(
        const float* __restrict__ Hf, const float* __restrict__ a_src,
        const float* __restrict__ a_dst, float* __restrict__ als,
        float* __restrict__ ald) {
    __shared__ float s1[256], s2[256];
    const int n = blockIdx.x;
    const int t = threadIdx.x;
    float v = Hf[(size_t)n * HC + t];
    s1[t] = v * a_src[t];
    s2[t] = v * a_dst[t];
    __syncthreads();
    #pragma unroll
    for (int off = 32; off >= 1; off >>= 1) {
        if ((t & 63) < off) { s1[t] += s1[t + off]; s2[t] += s2[t + off]; }
        __syncthreads();
    }
    if ((t & 63) == 0) {
        int h = t >> 6;
        als[(size_t)n * NH + h] = s1[t];
        ald[(size_t)n * NH + h] = s2[t];
    }
}

// ---------------- edge logits + segment max ----------------
__global__ void gat_edge_logits(const int* __restrict__ src, const int* __restrict__ dst,
        const float* __restrict__ als, const float* __restrict__ ald,
        float* __restrict__ elog, unsigned* __restrict__ menc, int E, int Etot) {
    int e = blockIdx.x * blockDim.x + threadIdx.x;
    if (e >= Etot) return;
    int s = (e < E) ? src[e] : (e - E);
    int d = (e < E) ? dst[e] : (e - E);
    #pragma unroll
    for (int h = 0; h < NH; ++h) {
        float v = als[(size_t)s * NH + h] + ald[(size_t)d * NH + h];
        v = v > 0.0f ? v : NEG_SLOPE * v;
        elog[(size_t)e * NH + h] = v;
        atomicMax(&menc[(size_t)d * NH + h], enc_f(v));
    }
}

// ---------------- exp + segment sum ----------------
__global__ void gat_edge_expsum(const int* __restrict__ dst,
        float* __restrict__ elog, const unsigned* __restrict__ menc,
        float* __restrict__ z, int E, int Etot) {
    int e = blockIdx.x * blockDim.x + threadIdx.x;
    if (e >= Etot) return;
    int d = (e < E) ? dst[e] : (e - E);
    #pragma unroll
    for (int h = 0; h < NH; ++h) {
        float m  = dec_f(menc[(size_t)d * NH + h]);
        float ex = __expf(elog[(size_t)e * NH + h] - m);
        elog[(size_t)e * NH + h] = ex;
        atomicAdd(&z[(size_t)d * NH + h], ex);
    }
}

// ---------------- message gather / scatter-add: one block per edge ----------------
__global__ __launch_bounds__(256) void gat_edge_scatter(
        const int* __restrict__ src, const int* __restrict__ dst,
        const float* __restrict__ Hf, const float* __restrict__ ex,
        const float* __restrict__ z, float* __restrict__ outf, int E, int Etot) {
    int e = blockIdx.x;
    int t = threadIdx.x;           // head = t>>6, ch = t&63
    int s = (e < E) ? src[e] : (e - E);
    int d = (e < E) ? dst[e] : (e - E);
    int h = t >> 6;
    float alpha = ex[(size_t)e * NH + h] / (z[(size_t)d * NH + h] + 1e-16f);
    float val = Hf[(size_t)s * HC + t] * alpha;
    atomicAdd(&outf[(size_t)d * HC + t], val);
}

// ---------------- bias + relu (layers 1,2, in place on [N,256]) ----------------
__global__ void gat_bias_relu(float* __restrict__ io, const float* __restrict__ b, size_t n) {
    size_t i = (size_t)blockIdx.x * blockDim.x + threadIdx.x;
    if (i >= n) return;
    float v = io[i] + b[i & (HC - 1)];
    io[i] = v > 0.0f ? v : 0.0f;
}

// ---------------- layer3: mean over heads + bias + relu -> [N,64] ----------------
__global__ void gat_head_mean(const float* __restrict__ agg, const float* __restrict__ b3,
                              float* __restrict__ x3, int N) {
    int i = blockIdx.x * blockDim.x + threadIdx.x;
    if (i >= N * CC) return;
    int n = i >> 6, c = i & (CC - 1);
    const float* r = agg + (size_t)n * HC;
    float v = 0.25f * (r[c] + r[CC + c] + r[2 * CC + c] + r[3 * CC + c]) + b3[c];
    x3[i] = v > 0.0f ? v : 0.0f;
}

// ---------------- global mean pool (sum + count) ----------------
__global__ void gat_pool_sum(const float* __restrict__ x3, const int* __restrict__ batch,
                             float* __restrict__ psum, float* __restrict__ pcnt, int N) {
    int i = blockIdx.x * blockDim.x + threadIdx.x;
    if (i >= N * CC) return;
    int n = i >> 6, c = i & (CC - 1);
    int g = batch[n];
    atomicAdd(&psum[(size_t)g * CC + c], x3[i]);
    if (c == 0) atomicAdd(&pcnt[g], 1.0f);
}

// ---------------- final linear: [G,64] @ [64,10] + bl ----------------
__global__ __launch_bounds__(64) void gat_final_linear(
        const float* __restrict__ psum, const float* __restrict__ pcnt,
        const float* __restrict__ Wl, const float* __restrict__ bl,
        float* __restrict__ out, int NC) {
    __shared__ float p[CC];
    int g = blockIdx.x, t = threadIdx.x;
    float cnt = pcnt[g]; cnt = cnt > 1.0f ? cnt : 1.0f;
    p[t] = psum[(size_t)g * CC + t] / cnt;
    __syncthreads();
    if (t < NC) {
        float acc = bl[t];
        #pragma unroll
        for (int c = 0; c < CC; ++c) acc += p[c] * Wl[c * NC + t];
        out[(size_t)g * NC + t] = acc;
    }
}

extern "C" void kernel_launch(void* const* d_in, const int* in_sizes, int n_in,
                              void* d_out, int out_size, void* d_ws, size_t ws_size,
                              hipStream_t stream) {
    const float* x    = (const float*)d_in[0];
    const int*   eidx = (const int*)d_in[1];
    const int*   batch= (const int*)d_in[2];
    const float* W1 = (const float*)d_in[3];
    const float* as1= (const float*)d_in[4];
    const float* ad1= (const float*)d_in[5];
    const float* b1 = (const float*)d_in[6];
    const float* W2 = (const float*)d_in[7];
    const float* as2= (const float*)d_in[8];
    const float* ad2= (const float*)d_in[9];
    const float* b2 = (const float*)d_in[10];
    const float* W3 = (const float*)d_in[11];
    const float* as3= (const float*)d_in[12];
    const float* ad3= (const float*)d_in[13];
    const float* b3 = (const float*)d_in[14];
    const float* Wl = (const float*)d_in[15];
    const float* bl = (const float*)d_in[16];

    const int F_IN = 128;
    const int NC   = 10;
    const int N    = in_sizes[0] / F_IN;
    const int E    = in_sizes[1] / 2;
    const int Etot = E + N;                 // with self loops
    const int G    = out_size / NC;
    const int* src = eidx;
    const int* dst = eidx + E;

    // ---- workspace carve-up ----
    float* p = (float*)d_ws;
    float*    bufH = p;               p += (size_t)N * HC;   // GEMM output h
    float*    bufX = p;               p += (size_t)N * HC;   // layer in/out (ping)
    float*    elog = p;               p += (size_t)Etot * NH;// edge logits/exp; reused as x3
    unsigned* menc = (unsigned*)p;    p += (size_t)N * NH;   // encoded segment max
    float*    zb   = p;               p += (size_t)N * NH;   // segment sum
    float*    als  = p;               p += (size_t)N * NH;
    float*    ald  = p;               p += (size_t)N * NH;
    float*    psum = p;               p += (size_t)G * CC;
    float*    pcnt = p;               p += (size_t)G;

    const int ntile = (N + 15) / 16;
    size_t nbig = (size_t)N * HC;

    auto layer = [&](const float* Xin, int K, const float* W,
                     const float* a_s, const float* a_d) {
        gat_gemm_wmma<<<dim3(ntile, HC / 128), 256, 0, stream>>>(Xin, W, bufH, K, HC, N);
        gat_attn_dots<<<N, 256, 0, stream>>>(bufH, a_s, a_d, als, ald);
        // zero aggregation target (bufX safe to reuse: GEMM already consumed it)
        gat_zero_f<<<(int)((nbig + 255) / 256), 256, 0, stream>>>(bufX, nbig);
        // menc and zb are contiguous: zero both (0u == encode-below-everything)
        gat_zero_f<<<(int)(((size_t)N * NH * 2 + 255) / 256), 256, 0, stream>>>(
            (float*)menc, (size_t)N * NH * 2);
        gat_edge_logits<<<(Etot + 255) / 256, 256, 0, stream>>>(src, dst, als, ald,
                                                               elog, menc, E, Etot);
        gat_edge_expsum<<<(Etot + 255) / 256, 256, 0, stream>>>(dst, elog, menc, zb,
                                                                E, Etot);
        gat_edge_scatter<<<Etot, 256, 0, stream>>>(src, dst, bufH, elog, zb, bufX,
                                                   E, Etot);
    };

    // layer 1: x[N,128] -> bufX[N,256]
    layer(x, F_IN, W1, as1, ad1);
    gat_bias_relu<<<(int)((nbig + 255) / 256), 256, 0, stream>>>(bufX, b1, nbig);
    // layer 2: bufX -> bufX
    layer(bufX, HC, W2, as2, ad2);
    gat_bias_relu<<<(int)((nbig + 255) / 256), 256, 0, stream>>>(bufX, b2, nbig);
    // layer 3: bufX -> agg in bufX, then head-mean into x3 (reuse elog region)
    layer(bufX, HC, W3, as3, ad3);
    float* x3 = elog;
    gat_head_mean<<<(N * CC + 255) / 256, 256, 0, stream>>>(bufX, b3, x3, N);

    // global mean pool + final linear
    gat_zero_f<<<(int)(((size_t)G * CC + G + 255) / 256), 256, 0, stream>>>(
        psum, (size_t)G * CC + G);
    gat_pool_sum<<<(N * CC + 255) / 256, 256, 0, stream>>>(x3, batch, psum, pcnt, N);
    gat_final_linear<<<G, 64, 0, stream>>>(psum, pcnt, Wl, bl, (float*)d_out, NC);
}